// MyDecoder_36687610642729
// MI455X (gfx1250) — compile-verified
//
#include <hip/hip_runtime.h>
#include <math.h>

#define T_STEPS 128
#define HID     512
#define NENC    256
#define VOCAB   32000
#define G3      1536          // 3*HID
#define PADLD   516           // LDS row stride (floats): 516 % 64 = 4 -> conflict-free frag reads

typedef float v2f __attribute__((ext_vector_type(2)));
typedef float v8f __attribute__((ext_vector_type(8)));

// -------------------------------------------------------------------------
// WMMA GEMM:  C[128 x N] = A[128 x 512] * W^T + bias,  W is N x K row-major
// with row stride ldw.  If idx != nullptr, A row t = Arows[idx[t]].
// One block = 256 threads = 8 waves; block covers 256 columns (32 per wave,
// two 16-col B fragments so each LDS A-fragment feeds two WMMAs).
// Per wave: 16 accumulator tiles (full M=128), K swept 4 at a time with
// v_wmma_f32_16x16x4_f32.  A panel staged once in LDS (264 KB).
// -------------------------------------------------------------------------
__global__ __launch_bounds__(256) void wmma_gemm128(
    const float* __restrict__ Arows, const int* __restrict__ idx,
    const float* __restrict__ W, int ldw,
    const float* __restrict__ bias,
    float* __restrict__ C, int ldc)
{
    __shared__ float Ash[128 * PADLD];          // 264 KB of the 320 KB WGP LDS

    // ---- stage A panel (128 x 512 fp32) into LDS, float4 at a time ----
    for (int i = threadIdx.x; i < 128 * 128; i += 256) {
        int row = i >> 7;
        int c4  = i & 127;
        const float* src = (idx ? (Arows + (size_t)idx[row] * HID)
                                : (Arows + (size_t)row   * HID)) + c4 * 4;
        float4 v = *(const float4*)src;
        *(float4*)&Ash[row * PADLD + c4 * 4] = v;
    }
    __syncthreads();

    const int lane   = threadIdx.x & 31;
    const int wave   = threadIdx.x >> 5;                 // 0..7
    const int n0     = blockIdx.x * 256 + wave * 32 + (lane & 15);  // strip 0
    const int n1     = n0 + 16;                                      // strip 1
    const int ksel   = (lane >> 4) * 2;                  // lanes 16-31 hold K+2
    const int m_lane = lane & 15;

    v8f acc[2][8];
    #pragma unroll
    for (int s = 0; s < 2; ++s)
        #pragma unroll
        for (int mt = 0; mt < 8; ++mt) acc[s][mt] = (v8f)0.0f;

    const float* wrow0 = W + (size_t)n0 * ldw;
    const float* wrow1 = W + (size_t)n1 * ldw;

    for (int k = 0; k < HID; k += 4) {
        const int k0 = k + ksel;
        if ((k & 63) == 0) {                              // keep W streams ahead
            __builtin_prefetch(wrow0 + k0 + 256, 0, 3);
            __builtin_prefetch(wrow1 + k0 + 256, 0, 3);
        }
        v2f b0 = *(const v2f*)(wrow0 + k0);               // W[n0][k0..k0+1]
        v2f b1 = *(const v2f*)(wrow1 + k0);               // W[n1][k0..k0+1]
        #pragma unroll
        for (int mt = 0; mt < 8; ++mt) {
            v2f a = *(const v2f*)&Ash[(mt * 16 + m_lane) * PADLD + k0];
            acc[0][mt] = __builtin_amdgcn_wmma_f32_16x16x4_f32(
                             false, a, false, b0, (short)0, acc[0][mt], false, false);
            acc[1][mt] = __builtin_amdgcn_wmma_f32_16x16x4_f32(
                             false, a, false, b1, (short)0, acc[1][mt], false, false);
        }
    }

    const float bv0  = bias ? bias[n0] : 0.0f;
    const float bv1  = bias ? bias[n1] : 0.0f;
    const int   rsel = (lane >> 4) * 8;                  // lanes 16-31: rows M+8
    #pragma unroll
    for (int mt = 0; mt < 8; ++mt) {
        #pragma unroll
        for (int j = 0; j < 8; ++j) {
            int row = mt * 16 + rsel + j;
            C[(size_t)row * ldc + n0] = acc[0][mt][j] + bv0;
            C[(size_t)row * ldc + n1] = acc[1][mt][j] + bv1;
        }
    }
}

// -------------------------------------------------------------------------
// Sequential recurrence: attention softmax + GRU, 128 steps.
// One workgroup, 1024 threads (32 waves).  Uses precomputed
// Gx[t] = W_ih[:, :512] @ emb[x[t]] + b_ih.
// -------------------------------------------------------------------------
__device__ inline float wave_sum(float v) {
    #pragma unroll
    for (int off = 16; off; off >>= 1) v += __shfl_xor(v, off, 32);
    return v;
}
__device__ inline float wave_max(float v) {
    #pragma unroll
    for (int off = 16; off; off >>= 1) v = fmaxf(v, __shfl_xor(v, off, 32));
    return v;
}

__global__ __launch_bounds__(1024) void gru_attn_steps(
    const float* __restrict__ h0, const float* __restrict__ enc,
    const float* __restrict__ Wih, const float* __restrict__ Whh,
    const float* __restrict__ bhh, const float* __restrict__ Gx,
    float* __restrict__ Hout, float* __restrict__ outW,
    float* __restrict__ hidden_out)
{
    __shared__ float h_s[HID];
    __shared__ float attn_s[HID];
    __shared__ float sc[NENC];      // scores, then softmax weights
    __shared__ float gi_s[G3];
    __shared__ float gh_s[G3];

    const int tid  = threadIdx.x;
    const int lane = tid & 31;
    const int wv   = tid >> 5;      // 0..31

    if (tid < HID) h_s[tid] = h0[tid];

    for (int t = 0; t < T_STEPS; ++t) {
        __syncthreads();
        // ---- scores[i] = enc[i] . h   (8 rows per wave) ----
        for (int i = wv; i < NENC; i += 32) {
            const float* er = enc + (size_t)i * HID;
            float s = 0.0f;
            #pragma unroll 4
            for (int j = lane; j < HID; j += 32) s += er[j] * h_s[j];
            s = wave_sum(s);
            if (lane == 0) sc[i] = s;
        }
        __syncthreads();
        // ---- softmax over 256 scores (wave 0) ----
        if (wv == 0) {
            float mx = -3.402823466e38f;
            for (int i = lane; i < NENC; i += 32) mx = fmaxf(mx, sc[i]);
            mx = wave_max(mx);
            float ex[8]; int c = 0; float sum = 0.0f;
            for (int i = lane; i < NENC; i += 32) { float e = __expf(sc[i] - mx); ex[c++] = e; sum += e; }
            sum = wave_sum(sum);
            const float inv = 1.0f / sum; c = 0;
            for (int i = lane; i < NENC; i += 32) {
                float wgt = ex[c++] * inv;
                sc[i] = wgt;
                outW[(size_t)t * NENC + i] = wgt;
            }
        }
        __syncthreads();
        // ---- attn[j] = sum_i w[i] * enc[i][j]  (coalesced over j) ----
        if (tid < HID) {
            float a = 0.0f;
            #pragma unroll 4
            for (int i = 0; i < NENC; ++i) a += sc[i] * enc[(size_t)i * HID + tid];
            attn_s[tid] = a;
        }
        __syncthreads();
        // ---- gi[r] = Gx[t][r] + W_ih[r,512:] . attn ;  gh[r] = W_hh[r] . h + b_hh ----
        for (int r = wv; r < G3; r += 32) {
            const float* wa = Wih + (size_t)r * (2 * HID) + HID;   // attn half
            const float* wh = Whh + (size_t)r * HID;
            float s1 = 0.0f, s2 = 0.0f;
            #pragma unroll 4
            for (int j = lane; j < HID; j += 32) { s1 += wa[j] * attn_s[j]; s2 += wh[j] * h_s[j]; }
            s1 = wave_sum(s1);
            s2 = wave_sum(s2);
            if (lane == 0) {
                gi_s[r] = Gx[(size_t)t * G3 + r] + s1;
                gh_s[r] = s2 + bhh[r];
            }
        }
        __syncthreads();
        // ---- GRU gates + state update ----
        if (tid < HID) {
            const int j = tid;
            float rg = 1.0f / (1.0f + __expf(-(gi_s[j]       + gh_s[j])));
            float zg = 1.0f / (1.0f + __expf(-(gi_s[HID + j] + gh_s[HID + j])));
            float ng = tanhf(gi_s[2 * HID + j] + rg * gh_s[2 * HID + j]);
            float hn = (1.0f - zg) * ng + zg * h_s[j];
            Hout[(size_t)t * HID + j] = hn;
            h_s[j] = hn;
        }
    }
    __syncthreads();
    if (tid < HID) hidden_out[tid] = h_s[tid];
}

// -------------------------------------------------------------------------
extern "C" void kernel_launch(void* const* d_in, const int* in_sizes, int n_in,
                              void* d_out, int out_size, void* d_ws, size_t ws_size,
                              hipStream_t stream) {
    const int*   x   = (const int*)  d_in[0];
    const float* h0  = (const float*)d_in[1];
    const float* enc = (const float*)d_in[2];
    const float* emb = (const float*)d_in[3];
    const float* Wih = (const float*)d_in[4];
    const float* Whh = (const float*)d_in[5];
    const float* bih = (const float*)d_in[6];
    const float* bhh = (const float*)d_in[7];
    const float* Wd  = (const float*)d_in[8];
    const float* bd  = (const float*)d_in[9];

    float* out    = (float*)d_out;
    float* logits = out;                                  // (128, 1, 32000)
    float* hid    = out + (size_t)T_STEPS * VOCAB;        // (1, 1, 512)
    float* wout   = hid + HID;                            // (128, 256)

    float* Gx   = (float*)d_ws;                           // 128 x 1536
    float* Hout = Gx + (size_t)T_STEPS * G3;              // 128 x 512

    // Phase 0: Gx = Emb[x] @ W_ih[:, :512]^T + b_ih   (N=1536 -> 6 blocks)
    wmma_gemm128<<<dim3(G3 / 256), dim3(256), 0, stream>>>(
        emb, x, Wih, 2 * HID, bih, Gx, G3);

    // Phase 1: sequential attention + GRU (single persistent workgroup)
    gru_attn_steps<<<dim3(1), dim3(1024), 0, stream>>>(
        h0, enc, Wih, Whh, bhh, Gx, Hout, wout, hid);

    // Phase 2: logits = H @ W_dense^T + b_dense   (N=32000 -> 125 blocks)
    wmma_gemm128<<<dim3(VOCAB / 256), dim3(256), 0, stream>>>(
        Hout, nullptr, Wd, HID, bd, logits, VOCAB);
}